// SheafConvLayer_9302899163386
// MI455X (gfx1250) — compile-verified
//
#include <hip/hip_runtime.h>
#include <hip/hip_bf16.h>

// ---------------------------------------------------------------------------
// SheafConvLayer for MI455X (gfx1250, wave32).
//   out = x - STEP * (diag[:,None]*y + scatter_add(norm_maps*y[col] -> row))
//   y   = x @ W_lin^T + b_lin        (fp32 WMMA 16x16x4, 64x64 reg-blocked)
// ---------------------------------------------------------------------------

typedef __attribute__((ext_vector_type(2))) float v2f;
typedef __attribute__((ext_vector_type(8))) float v8f;

#define STEP_SIZE 0.5f

__device__ __forceinline__ void atomic_add_f32(float* p, float v) {
  // lowers to global_atomic_add_f32 (no CAS loop)
  unsafeAtomicAdd(p, v);
}

// ---------------------------------------------------------------------------
// 1) Per-node dual dot products: s[i] = x[i,:].w[:D], t[i] = x[i,:].w[D:2D]
//    One wave (32 lanes) per node, float4 loads, shfl_xor wave32 reduction.
// ---------------------------------------------------------------------------
__global__ __launch_bounds__(256) void k_node_st(
    const float* __restrict__ x, const float* __restrict__ wsheaf,
    float* __restrict__ s, float* __restrict__ t, int N, int D) {
  int wid  = (blockIdx.x * blockDim.x + threadIdx.x) >> 5;
  int lane = threadIdx.x & 31;
  if (wid >= N) return;
  const float4* xr = (const float4*)(x + (size_t)wid * D);
  const float4* w0 = (const float4*)(wsheaf);
  const float4* w1 = (const float4*)(wsheaf + D);
  int d4 = D >> 2;
  float as = 0.f, at = 0.f;
  for (int k = lane; k < d4; k += 32) {
    float4 xv = xr[k];
    float4 a  = w0[k];
    float4 b  = w1[k];
    as += xv.x * a.x + xv.y * a.y + xv.z * a.z + xv.w * a.w;
    at += xv.x * b.x + xv.y * b.y + xv.z * b.z + xv.w * b.w;
  }
  #pragma unroll
  for (int off = 16; off > 0; off >>= 1) {
    as += __shfl_xor(as, off, 32);
    at += __shfl_xor(at, off, 32);
  }
  if (lane == 0) { s[wid] = as; t[wid] = at; }
}

// ---------------------------------------------------------------------------
// 2) Zero the per-node diagonal accumulator (workspace is not pre-zeroed).
// ---------------------------------------------------------------------------
__global__ void k_zero(float* __restrict__ p, int n) {
  int i = blockIdx.x * blockDim.x + threadIdx.x;
  if (i < n) p[i] = 0.f;
}

// ---------------------------------------------------------------------------
// 3) Per-edge restriction map + diag segment-sum:
//    maps[e] = tanh(s[row]+t[col]);  diag_maps[row] += maps^2 (atomic)
// ---------------------------------------------------------------------------
__global__ void k_edge_maps(
    const int* __restrict__ row, const int* __restrict__ col,
    const float* __restrict__ s, const float* __restrict__ t,
    float* __restrict__ maps, float* __restrict__ diag_maps, int E) {
  int e = blockIdx.x * blockDim.x + threadIdx.x;
  if (e >= E) return;
  int r = row[e], c = col[e];
  float m = tanhf(s[r] + t[c]);
  maps[e] = m;
  atomic_add_f32(diag_maps + r, m * m);
}

// ---------------------------------------------------------------------------
// 4) Per-node normalization: d_inv = (diag_maps+1)^-1/2,
//    sd = STEP * d_inv * diag_maps * d_inv   (folded scale for residual)
// ---------------------------------------------------------------------------
__global__ void k_node_dinv(const float* __restrict__ diag_maps,
                            float* __restrict__ d_inv, float* __restrict__ sd,
                            int N) {
  int i = blockIdx.x * blockDim.x + threadIdx.x;
  if (i >= N) return;
  float dm = diag_maps[i];
  float di = rsqrtf(dm + 1.0f);
  d_inv[i] = di;
  sd[i]    = STEP_SIZE * di * dm * di;
}

// ---------------------------------------------------------------------------
// 5) Per-edge coefficient.  norm = d_inv[row]*(-m*m_rev)*d_inv[col];
//    out contribution is  -STEP*norm*y[col]  =>
//    coef[e] = +STEP * d_inv[row]*maps[e]*maps[rev[e]]*d_inv[col]
// ---------------------------------------------------------------------------
__global__ void k_edge_coef(
    const int* __restrict__ row, const int* __restrict__ col,
    const int* __restrict__ rev, const float* __restrict__ maps,
    const float* __restrict__ d_inv, float* __restrict__ coef, int E) {
  int e = blockIdx.x * blockDim.x + threadIdx.x;
  if (e >= E) return;
  coef[e] = STEP_SIZE * d_inv[row[e]] * maps[e] * maps[rev[e]] * d_inv[col[e]];
}

// ---------------------------------------------------------------------------
// 6) GEMM: y[i,j] = sum_k x[i,k]*W[j,k] + b[j]  via V_WMMA_F32_16X16X4_F32.
//    Register-blocked: one wave computes a 64x64 output tile = 4x4 grid of
//    16x16 WMMA tiles (16 accumulators, 128 VGPRs).  Per K-step of 4:
//    4 A-frag + 4 B-frag b64 loads feed 16 WMMAs (2048 FLOP per 8B/lane).
//    B[k,n] = W[n,k], so reading W row-major with the A addressing pattern
//    yields the B fragment directly (no transpose).
//    Mapping: blockIdx.x = M-tile, wave-in-block = N-tile (needs D == 512).
// ---------------------------------------------------------------------------
__global__ __launch_bounds__(256) void k_gemm_wmma(
    const float* __restrict__ x, const float* __restrict__ W,
    const float* __restrict__ b, float* __restrict__ y, int N, int D) {
  const int lane  = threadIdx.x & 31;
  const int tileM = blockIdx.x;             // 64 rows of y
  const int tileN = threadIdx.x >> 5;       // 64 cols of y (8 waves = 512/64)
  if (tileN * 64 >= D) return;              // wave-uniform

  const int m    = lane & 15;               // row within 16-subtile / B col
  const int half = lane >> 4;               // K sub-pair select (K+0,1 / K+2,3)

  // A sub-tile base pointers (clamped for the ragged last M-tile; stores
  // are guarded so clamped rows never write).
  const float* ap[4];
  #pragma unroll
  for (int i = 0; i < 4; ++i) {
    int r = tileM * 64 + i * 16 + m;
    if (r >= N) r = N - 1;
    ap[i] = x + (size_t)r * D + half * 2;
  }
  // B base pointer; j sub-tile offset is j*16*D floats = 32KB (24-bit imm).
  const float* bp = W + (size_t)(tileN * 64 + m) * D + half * 2;

  v8f acc[4][4] = {};
  #pragma unroll 2
  for (int k = 0; k < D; k += 4) {
    v2f a[4], bb[4];
    #pragma unroll
    for (int i = 0; i < 4; ++i) a[i]  = *(const v2f*)(ap[i] + k);
    #pragma unroll
    for (int j = 0; j < 4; ++j) bb[j] = *(const v2f*)(bp + j * 16 * D + k);
    #pragma unroll
    for (int i = 0; i < 4; ++i)
      #pragma unroll
      for (int j = 0; j < 4; ++j)
        acc[i][j] = __builtin_amdgcn_wmma_f32_16x16x4_f32(
            false, a[i], false, bb[j], (short)0, acc[i][j], false, false);
  }

  // C/D layout: VGPR r -> M = r (lanes 0-15) or r+8 (lanes 16-31), N = lane&15
  #pragma unroll
  for (int j = 0; j < 4; ++j) {
    int colBase = tileN * 64 + j * 16 + m;
    float bias  = b[colBase];
    #pragma unroll
    for (int i = 0; i < 4; ++i) {
      #pragma unroll
      for (int r = 0; r < 8; ++r) {
        int rr = tileM * 64 + i * 16 + r + half * 8;
        if (rr < N) y[(size_t)rr * D + colBase] = acc[i][j][r] + bias;
      }
    }
  }
}

// ---------------------------------------------------------------------------
// 7) Residual + diagonal term: out = x - (STEP*diag) * y   (float4)
// ---------------------------------------------------------------------------
__global__ void k_out_init(const float* __restrict__ x,
                           const float* __restrict__ y,
                           const float* __restrict__ sd,
                           float* __restrict__ out, int total4, int d4) {
  int i = blockIdx.x * blockDim.x + threadIdx.x;
  if (i >= total4) return;
  int node = i / d4;
  float4 xv = ((const float4*)x)[i];
  float4 yv = ((const float4*)y)[i];
  float  sc = sd[node];
  float4 o;
  o.x = xv.x - sc * yv.x;
  o.y = xv.y - sc * yv.y;
  o.z = xv.z - sc * yv.z;
  o.w = xv.w - sc * yv.w;
  ((float4*)out)[i] = o;
}

// ---------------------------------------------------------------------------
// 8) Edge scatter: out[row,:] += coef[e] * y[col,:]   (block per edge,
//    float4 gather from L2-resident y, hw fp32 atomics into out; each of the
//    128 threads handles exactly one float4 of the 512-wide row)
// ---------------------------------------------------------------------------
__global__ __launch_bounds__(128) void k_edge_scatter(
    const int* __restrict__ row, const int* __restrict__ col,
    const float* __restrict__ coef, const float* __restrict__ y,
    float* __restrict__ out, int E, int D) {
  int e = blockIdx.x;
  if (e >= E) return;
  float  kc = coef[e];
  size_t rb = (size_t)row[e] * D;
  size_t cb = (size_t)col[e] * D;
  for (int j = (threadIdx.x << 2); j < D; j += (blockDim.x << 2)) {
    float4 yv = *(const float4*)(y + cb + j);
    atomic_add_f32(out + rb + j + 0, kc * yv.x);
    atomic_add_f32(out + rb + j + 1, kc * yv.y);
    atomic_add_f32(out + rb + j + 2, kc * yv.z);
    atomic_add_f32(out + rb + j + 3, kc * yv.w);
  }
}

// ---------------------------------------------------------------------------
extern "C" void kernel_launch(void* const* d_in, const int* in_sizes, int n_in,
                              void* d_out, int out_size, void* d_ws, size_t ws_size,
                              hipStream_t stream) {
  const float* x      = (const float*)d_in[0];
  const float* W_lin  = (const float*)d_in[1];
  const float* b_lin  = (const float*)d_in[2];
  const float* wsheaf = (const float*)d_in[3];
  const int*   ei     = (const int*)d_in[4];
  const int*   rev    = (const int*)d_in[5];

  const int D = in_sizes[2];          // 512
  const int N = in_sizes[0] / D;      // 50000
  const int E = in_sizes[5];          // 200000
  const int* row = ei;
  const int* col = ei + E;

  // workspace carve-up (all fp32)
  float* y    = (float*)d_ws;             // N*D
  float* s    = y + (size_t)N * D;        // N
  float* t    = s + N;                    // N
  float* maps = t + N;                    // E
  float* dm   = maps + E;                 // N
  float* dinv = dm + N;                   // N
  float* sd   = dinv + N;                 // N
  float* coef = sd + N;                   // E

  float* out = (float*)d_out;

  // 1) s,t per node (one wave per node)
  k_node_st<<<(N + 7) / 8, 256, 0, stream>>>(x, wsheaf, s, t, N, D);
  // 2) zero diag accumulator
  k_zero<<<(N + 255) / 256, 256, 0, stream>>>(dm, N);
  // 3) maps + diag segment sum
  k_edge_maps<<<(E + 255) / 256, 256, 0, stream>>>(row, col, s, t, maps, dm, E);
  // 4) d_inv / scaled diag
  k_node_dinv<<<(N + 255) / 256, 256, 0, stream>>>(dm, dinv, sd, N);
  // 5) edge coefficients
  k_edge_coef<<<(E + 255) / 256, 256, 0, stream>>>(row, col, rev, maps, dinv, coef, E);
  // 6) WMMA GEMM y = x @ W^T + b  (block = 64-row M-tile, 8 waves = N-tiles)
  int tilesM = (N + 63) / 64;
  k_gemm_wmma<<<tilesM, 256, 0, stream>>>(x, W_lin, b_lin, y, N, D);
  // 7) out = x - STEP*diag*y
  int total4 = (N * D) / 4;
  k_out_init<<<(total4 + 255) / 256, 256, 0, stream>>>(x, y, sd, out, total4, D / 4);
  // 8) edge scatter with fp32 atomics
  k_edge_scatter<<<E, 128, 0, stream>>>(row, col, coef, y, out, E, D);
}